// PhotoMetricLoss_4501125726406
// MI455X (gfx1250) — compile-verified
//
#include <hip/hip_runtime.h>
#include <hip/hip_bf16.h>

typedef __attribute__((ext_vector_type(2))) float v2f;
typedef __attribute__((ext_vector_type(8))) float v8f;

constexpr int Bn = 8;
constexpr int Cn = 3;
constexpr int Hn = 512;
constexpr int Wn = 1024;
constexpr int HW = Hn * Wn;
constexpr float FLOW_COEF = 20.0f;

// Full 32-lane sum using V_WMMA_F32_16X16X4_F32.
// A (16x4) carries the per-lane value in its first VGPR and 0 in the second;
// B (4x16) is all ones. Then for any column n: sum_m D[m][n] = sum of ALL
// elements of A = sum over the 32 lane values -- independent of the exact
// A/B lane striping. Each lane holds D rows (0..7 | 8..15) for one column in
// its 8 C VGPRs, so (sum of own 8) + (partner via shfl_xor 16) = wave total.
__device__ __forceinline__ float wave_sum32(float v) {
  v2f a; a.x = v;    a.y = 0.0f;
  v2f b; b.x = 1.0f; b.y = 1.0f;
  v8f c = {};
  c = __builtin_amdgcn_wmma_f32_16x16x4_f32(
      /*neg_a=*/false, a, /*neg_b=*/false, b,
      /*c_mod=*/(short)0, c, /*reuse_a=*/false, /*reuse_b=*/false);
  float s = c[0] + c[1] + c[2] + c[3] + c[4] + c[5] + c[6] + c[7];
  s += __shfl_xor(s, 16, 32);
  return s;
}

// Sum over channels of (bilinear_warp(src, 20*flow)[p] - tgt[p])^2.
// Zero-padding is folded into the tap weights (identical to the reference's
// gather-with-valid-mask formulation since g*valid*w == (w*valid)*src[clamped]).
__device__ __forceinline__ float warp_sqdiff(const float* __restrict__ src,
                                             const float* __restrict__ tgt,
                                             const float* __restrict__ flow,
                                             int p, int x, int y) {
  const float fx = __fmaf_rn(FLOW_COEF, flow[p],      (float)x);
  const float fy = __fmaf_rn(FLOW_COEF, flow[HW + p], (float)y);
  const float x0f = floorf(fx);
  const float y0f = floorf(fy);
  const float wx = fx - x0f;
  const float wy = fy - y0f;
  const int x0 = (int)x0f, y0 = (int)y0f;
  const int x1 = x0 + 1,   y1 = y0 + 1;

  const float vx0 = (x0 >= 0 && x0 <= Wn - 1) ? 1.0f : 0.0f;
  const float vx1 = (x1 >= 0 && x1 <= Wn - 1) ? 1.0f : 0.0f;
  const float vy0 = (y0 >= 0 && y0 <= Hn - 1) ? 1.0f : 0.0f;
  const float vy1 = (y1 >= 0 && y1 <= Hn - 1) ? 1.0f : 0.0f;

  const int cx0 = min(max(x0, 0), Wn - 1);
  const int cx1 = min(max(x1, 0), Wn - 1);
  const int cy0 = min(max(y0, 0), Hn - 1);
  const int cy1 = min(max(y1, 0), Hn - 1);

  const int i00 = cy0 * Wn + cx0;
  const int i10 = cy0 * Wn + cx1;
  const int i01 = cy1 * Wn + cx0;
  const int i11 = cy1 * Wn + cx1;

  const float w00 = (1.0f - wx) * (1.0f - wy) * vx0 * vy0;
  const float w10 = wx * (1.0f - wy) * vx1 * vy0;
  const float w01 = (1.0f - wx) * wy * vx0 * vy1;
  const float w11 = wx * wy * vx1 * vy1;

  float s = 0.0f;
#pragma unroll
  for (int ch = 0; ch < Cn; ++ch) {
    const float* __restrict__ sp = src + (size_t)ch * HW;
    float g = w00 * sp[i00];
    g = __fmaf_rn(w10, sp[i10], g);
    g = __fmaf_rn(w01, sp[i01], g);
    g = __fmaf_rn(w11, sp[i11], g);
    const float d = g - tgt[(size_t)ch * HW + p];
    s = __fmaf_rn(d, d, s);
  }
  return s;
}

__global__ void zero_acc_kernel(float* __restrict__ acc) {
  if (threadIdx.x < 4 * Bn) acc[threadIdx.x] = 0.0f;
}

__global__ __launch_bounds__(256) void pm_partial_kernel(
    const float* __restrict__ recon_f, const float* __restrict__ recon_b,
    const float* __restrict__ flow_f,  const float* __restrict__ flow_b,
    const float* __restrict__ occ_f,   const float* __restrict__ occ_b,
    float* __restrict__ acc) {
  const int p = blockIdx.x * blockDim.x + threadIdx.x;  // pixel index in H*W
  const int b = blockIdx.y;
  const int x = p & (Wn - 1);
  const int y = p >> 10;  // Wn == 1024

  const float* __restrict__ rf = recon_f + (size_t)b * Cn * HW;
  const float* __restrict__ rb = recon_b + (size_t)b * Cn * HW;
  const float* __restrict__ ff = flow_f  + (size_t)b * 2 * HW;
  const float* __restrict__ fb = flow_b  + (size_t)b * 2 * HW;

  // forward: warp recon_b by flow_f, compare against recon_f, mask by occ_f
  const float sq_f = warp_sqdiff(rb, rf, ff, p, x, y);
  const float ofv  = occ_f[(size_t)b * HW + p];
  const float numf = sq_f * (1.0f - ofv);

  // backward: warp recon_f by flow_b, compare against recon_b, mask by occ_b
  const float sq_b = warp_sqdiff(rf, rb, fb, p, x, y);
  const float obv  = occ_b[(size_t)b * HW + p];
  const float numb = sq_b * (1.0f - obv);

  // Wave-level reduction on the matrix pipe (EXEC is all-1s here: full blocks,
  // no divergence before this point).
  const float t0 = wave_sum32(numf);
  const float t1 = wave_sum32(ofv);
  const float t2 = wave_sum32(numb);
  const float t3 = wave_sum32(obv);

  __shared__ float lds[8 * 4];  // 8 waves x 4 accumulators
  const int wave = threadIdx.x >> 5;
  const int lane = threadIdx.x & 31;
  if (lane == 0) {
    lds[wave * 4 + 0] = t0;
    lds[wave * 4 + 1] = t1;
    lds[wave * 4 + 2] = t2;
    lds[wave * 4 + 3] = t3;
  }
  __syncthreads();
  if (threadIdx.x < 4) {
    float s = 0.0f;
#pragma unroll
    for (int w = 0; w < 8; ++w) s += lds[w * 4 + threadIdx.x];
    atomicAdd(&acc[b * 4 + threadIdx.x], s);  // native global_atomic_add_f32
  }
}

__global__ void finalize_kernel(const float* __restrict__ acc,
                                float* __restrict__ out) {
  if (threadIdx.x == 0 && blockIdx.x == 0) {
    const float HWf = (float)HW;
    float loss = 0.0f;
    for (int b = 0; b < Bn; ++b) {
      const float nf  = acc[b * 4 + 0];
      const float sof = acc[b * 4 + 1];
      const float nb  = acc[b * 4 + 2];
      const float sob = acc[b * 4 + 3];
      loss += nf / ((float)Cn * (HWf - sof));
      loss += nb / ((float)Cn * (HWf - sob));
    }
    out[0] = loss / (float)Bn;
  }
}

extern "C" void kernel_launch(void* const* d_in, const int* in_sizes, int n_in,
                              void* d_out, int out_size, void* d_ws, size_t ws_size,
                              hipStream_t stream) {
  const float* recon_f = (const float*)d_in[0];
  const float* recon_b = (const float*)d_in[1];
  const float* flow_f  = (const float*)d_in[2];
  const float* flow_b  = (const float*)d_in[3];
  const float* occ_f   = (const float*)d_in[4];
  const float* occ_b   = (const float*)d_in[5];
  float* out = (float*)d_out;
  float* acc = (float*)d_ws;  // 32 floats of scratch

  zero_acc_kernel<<<1, 64, 0, stream>>>(acc);

  dim3 grid(HW / 256, Bn);
  pm_partial_kernel<<<grid, 256, 0, stream>>>(recon_f, recon_b, flow_f, flow_b,
                                              occ_f, occ_b, acc);

  finalize_kernel<<<1, 32, 0, stream>>>(acc, out);
}